// snowball_layer_16724602650837
// MI455X (gfx1250) — compile-verified
//
#include <hip/hip_runtime.h>

#define N_NODES 50000
#define N_EDGES 800000
#define IN_F 128
#define OUT_F 64
#define M_TILES (N_NODES / 16)   // 3125 exactly

typedef __attribute__((ext_vector_type(2))) float v2f;
typedef __attribute__((ext_vector_type(4))) float v4f;
typedef __attribute__((ext_vector_type(8))) float v8f;

// ---------------------------------------------------------------------------
// Kernel 1: XW = x @ W via V_WMMA_F32_16X16X4_F32 (exact f32 path).
// Block = 128 threads (4 waves). Each wave computes one 16-row strip of XW
// (M-tile) across all 4 N-tiles (OUT_F = 64 = 4 x 16), K-loop step 4.
// Weight matrix (128x64 f32 = 32 KB) is staged once into LDS per block.
// ---------------------------------------------------------------------------
__global__ __launch_bounds__(128) void gcn_gemm_xw(const float* __restrict__ x,
                                                   const float* __restrict__ w,
                                                   float* __restrict__ xw) {
    __shared__ float wlds[IN_F * OUT_F];  // 32 KB of the 320 KB WGP LDS

    {   // cooperative float4 copy of W into LDS
        const v4f* src = (const v4f*)w;
        v4f* dst = (v4f*)wlds;
        for (int i = threadIdx.x; i < (IN_F * OUT_F) / 4; i += blockDim.x)
            dst[i] = src[i];
    }
    __syncthreads();

    const int wave   = threadIdx.x >> 5;
    const int lane   = threadIdx.x & 31;
    const int half   = lane >> 4;   // 0: lanes 0-15, 1: lanes 16-31
    const int lane16 = lane & 15;
    const int mtile  = blockIdx.x * 4 + wave;
    if (mtile >= M_TILES) return;   // whole-wave uniform skip: EXEC stays all-1s for WMMA
    const int m0 = mtile * 16;

    // A-matrix 16x4 f32 layout (ISA 7.12.2): lanes 0-15 hold M=lane,K={k0,k0+1};
    // lanes 16-31 hold M=lane-16,K={k0+2,k0+3}.
    const float* arow = x + (size_t)(m0 + lane16) * IN_F;

    v8f acc0 = {}, acc1 = {}, acc2 = {}, acc3 = {};

    #pragma unroll 4
    for (int k0 = 0; k0 < IN_F; k0 += 4) {
        v2f a;
        a.x = arow[k0 + 2 * half + 0];
        a.y = arow[k0 + 2 * half + 1];

        // B-matrix 4x16 f32: lanes 0-15 -> rows K={k0,k0+1}, N=lane16;
        // lanes 16-31 -> rows K={k0+2,k0+3}, N=lane16 (mirrors A half-split).
        const int krow0 = (k0 + 2 * half) * OUT_F;
        const int krow1 = krow0 + OUT_F;
        v2f b0, b1, b2, b3;
        b0.x = wlds[krow0 +  0 + lane16]; b0.y = wlds[krow1 +  0 + lane16];
        b1.x = wlds[krow0 + 16 + lane16]; b1.y = wlds[krow1 + 16 + lane16];
        b2.x = wlds[krow0 + 32 + lane16]; b2.y = wlds[krow1 + 32 + lane16];
        b3.x = wlds[krow0 + 48 + lane16]; b3.y = wlds[krow1 + 48 + lane16];

        acc0 = __builtin_amdgcn_wmma_f32_16x16x4_f32(false, a, false, b0, (short)0, acc0, false, false);
        acc1 = __builtin_amdgcn_wmma_f32_16x16x4_f32(false, a, false, b1, (short)0, acc1, false, false);
        acc2 = __builtin_amdgcn_wmma_f32_16x16x4_f32(false, a, false, b2, (short)0, acc2, false, false);
        acc3 = __builtin_amdgcn_wmma_f32_16x16x4_f32(false, a, false, b3, (short)0, acc3, false, false);
    }

    // C/D 16x16 f32 layout: VGPR j, lanes 0-15 -> M=j, N=lane16; lanes 16-31 -> M=j+8.
    #pragma unroll
    for (int j = 0; j < 8; ++j) {
        float* o = xw + (size_t)(m0 + j + 8 * half) * OUT_F + lane16;
        o[ 0] = acc0[j];
        o[16] = acc1[j];
        o[32] = acc2[j];
        o[48] = acc3[j];
    }
}

// ---------------------------------------------------------------------------
// Kernel 2: out[n][f] = bias[f]  (harness poisons d_out; seed the accumulator)
// float2 stores, OUT_F=64 is even so pairs never straddle rows.
// ---------------------------------------------------------------------------
__global__ __launch_bounds__(256) void gcn_bias_init(const float* __restrict__ bias,
                                                     float* __restrict__ out) {
    const int i = blockIdx.x * blockDim.x + threadIdx.x;   // over float2 elements
    if (i >= (N_NODES * OUT_F) / 2) return;
    const int f = (i * 2) & (OUT_F - 1);
    float2 v;
    v.x = bias[f];
    v.y = bias[f + 1];
    ((float2*)out)[i] = v;
}

// ---------------------------------------------------------------------------
// Kernel 3: scatter-add. One wave32 per edge; each lane handles 2 features:
// coalesced float2 gather of XW[col] (L2-resident: XW = 12.8 MB << 192 MB L2),
// then two hardware f32 atomic adds into out[row] (L2 atomic units reduce).
// ---------------------------------------------------------------------------
__global__ __launch_bounds__(256) void gcn_edge_scatter(const float* __restrict__ xw,
                                                        const float* __restrict__ ew,
                                                        const int* __restrict__ er,
                                                        const int* __restrict__ ec,
                                                        float* __restrict__ out) {
    const int wave = threadIdx.x >> 5;
    const int lane = threadIdx.x & 31;
    const int e = blockIdx.x * 8 + wave;   // 8 waves (edges) per 256-thread block
    if (e >= N_EDGES) return;

    const int   r = er[e];
    const int   c = ec[e];
    const float wgt = ew[e];

    const float2 v = ((const float2*)(xw + (size_t)c * OUT_F))[lane];
    float* dst = out + (size_t)r * OUT_F + lane * 2;

    __hip_atomic_fetch_add(dst + 0, v.x * wgt, __ATOMIC_RELAXED, __HIP_MEMORY_SCOPE_AGENT);
    __hip_atomic_fetch_add(dst + 1, v.y * wgt, __ATOMIC_RELAXED, __HIP_MEMORY_SCOPE_AGENT);
}

// ---------------------------------------------------------------------------
extern "C" void kernel_launch(void* const* d_in, const int* in_sizes, int n_in,
                              void* d_out, int out_size, void* d_ws, size_t ws_size,
                              hipStream_t stream) {
    const float* x      = (const float*)d_in[0];
    const float* weight = (const float*)d_in[1];
    const float* bias   = (const float*)d_in[2];
    const float* ew     = (const float*)d_in[3];
    const int*   er     = (const int*)d_in[4];
    const int*   ec     = (const int*)d_in[5];
    float*       out    = (float*)d_out;
    float*       xw     = (float*)d_ws;   // needs N_NODES*OUT_F*4 = 12.8 MB scratch

    // GEMM: 3125 M-tiles, 4 waves per block
    gcn_gemm_xw<<<(M_TILES + 3) / 4, 128, 0, stream>>>(x, weight, xw);

    // Seed out with bias (independent of GEMM; stream order is fine)
    gcn_bias_init<<<((N_NODES * OUT_F / 2) + 255) / 256, 256, 0, stream>>>(bias, out);

    // Edge scatter: one wave per edge
    gcn_edge_scatter<<<(N_EDGES + 7) / 8, 256, 0, stream>>>(xw, ew, er, ec, out);
}